// CircuitLoss_6253472383136
// MI455X (gfx1250) — compile-verified
//
#include <hip/hip_runtime.h>
#include <hip/hip_bf16.h>
#include <math.h>

#define B_ROWS 8192
#define D_DIM  4096
#define K_CLS  64
#define EPSF   1e-8f

typedef __attribute__((ext_vector_type(2))) float v2f;
typedef __attribute__((ext_vector_type(4))) float f4;
typedef __attribute__((ext_vector_type(8))) float v8f;

// ---------------- labels + class counts from one-hot ----------------
__global__ void k_labels(const float* __restrict__ probs, int* __restrict__ labels,
                         float* __restrict__ counts) {
    int row = blockIdx.x * blockDim.x + threadIdx.x;
    if (row >= B_ROWS) return;
    int lab = 0;
    const float* pr = probs + (size_t)row * K_CLS;
    #pragma unroll 8
    for (int k = 0; k < K_CLS; ++k)
        if (pr[k] > 0.5f) lab = k;
    labels[row] = lab;
    atomicAdd(&counts[lab], 1.0f);
}

// ---------------- per-row norms: w[i]=1/max(||x||,eps), sumdiag[class]+=ssq*w^2 ----------------
__global__ void k_row_norms(const float* __restrict__ X, const int* __restrict__ labels,
                            float* __restrict__ w, float* __restrict__ sumdiag) {
    int row = blockIdx.x;
    const f4* xr = (const f4*)(X + (size_t)row * D_DIM);
    float ssq = 0.f;
    for (int i = threadIdx.x; i < D_DIM / 4; i += blockDim.x) {
        f4 v = xr[i];
        ssq += v.x * v.x + v.y * v.y + v.z * v.z + v.w * v.w;
    }
    __shared__ float red[256];
    red[threadIdx.x] = ssq;
    __syncthreads();
    for (int s = blockDim.x >> 1; s > 0; s >>= 1) {
        if (threadIdx.x < s) red[threadIdx.x] += red[threadIdx.x + s];
        __syncthreads();
    }
    if (threadIdx.x == 0) {
        float tot  = red[0];
        float inv  = 1.0f / fmaxf(sqrtf(tot), EPSF);
        w[row] = inv;
        atomicAdd(&sumdiag[labels[row]], tot * inv * inv);  // diag_i = ssq / max(||x||,eps)^2
    }
}

// ---------------- centroid GEMM:  C[64,D] += M^T * diag(w) * X  via V_WMMA_F32_16X16X4_F32 ----
// One wave per (col_tile, b_slice). 4 accumulator tiles cover all 64 classes.
#define NSLICE 32
__global__ void k_centroid_wmma(const float* __restrict__ X, const float* __restrict__ w,
                                const int* __restrict__ labels, float* __restrict__ C) {
    int wid  = (blockIdx.x * blockDim.x + threadIdx.x) >> 5;  // global wave id, 0..8191
    int lane = threadIdx.x & 31;
    int col_tile = wid >> 5;           // 0..255 (D/16 tiles)
    int slice    = wid & (NSLICE - 1); // 0..31
    int col0 = col_tile * 16;
    int rows_per_slice = B_ROWS / NSLICE;   // 256
    int b_begin = slice * rows_per_slice;
    int b_end   = b_begin + rows_per_slice;

    int m  = lane & 15;          // class-in-tile (A row) / col-in-tile (B col)
    int kh = (lane >> 4) << 1;   // K offset: lanes 0-15 -> K0/K1, lanes 16-31 -> K2/K3

    v8f acc0 = {}, acc1 = {}, acc2 = {}, acc3 = {};
    for (int b0 = b_begin; b0 < b_end; b0 += 4) {
        int   la = labels[b0 + kh];
        int   lb = labels[b0 + kh + 1];
        float wa = w[b0 + kh];
        float wb = w[b0 + kh + 1];
        v2f bm;
        bm.x = X[(size_t)(b0 + kh)     * D_DIM + col0 + m];
        bm.y = X[(size_t)(b0 + kh + 1) * D_DIM + col0 + m];
        v2f a0, a1, a2, a3;
        a0.x = (la == m      ) ? wa : 0.f;  a0.y = (lb == m      ) ? wb : 0.f;
        a1.x = (la == m + 16 ) ? wa : 0.f;  a1.y = (lb == m + 16 ) ? wb : 0.f;
        a2.x = (la == m + 32 ) ? wa : 0.f;  a2.y = (lb == m + 32 ) ? wb : 0.f;
        a3.x = (la == m + 48 ) ? wa : 0.f;  a3.y = (lb == m + 48 ) ? wb : 0.f;
        acc0 = __builtin_amdgcn_wmma_f32_16x16x4_f32(false, a0, false, bm, (short)0, acc0, false, false);
        acc1 = __builtin_amdgcn_wmma_f32_16x16x4_f32(false, a1, false, bm, (short)0, acc1, false, false);
        acc2 = __builtin_amdgcn_wmma_f32_16x16x4_f32(false, a2, false, bm, (short)0, acc2, false, false);
        acc3 = __builtin_amdgcn_wmma_f32_16x16x4_f32(false, a3, false, bm, (short)0, acc3, false, false);
    }
    // C/D layout: VGPR r, lanes 0-15 -> M=r, lanes 16-31 -> M=r+8 ; N = lane&15
    int mrow = (lane < 16) ? 0 : 8;
    int col  = col0 + (lane & 15);
    #pragma unroll
    for (int r = 0; r < 8; ++r) {
        atomicAdd(&C[(size_t)( 0 + r + mrow) * D_DIM + col], acc0[r]);
        atomicAdd(&C[(size_t)(16 + r + mrow) * D_DIM + col], acc1[r]);
        atomicAdd(&C[(size_t)(32 + r + mrow) * D_DIM + col], acc2[r]);
        atomicAdd(&C[(size_t)(48 + r + mrow) * D_DIM + col], acc3[r]);
    }
}

// ---------------- mSm[k] = ||C_k||^2 ----------------
__global__ void k_class_sqnorm(const float* __restrict__ C, float* __restrict__ mSm) {
    int k = blockIdx.x;
    float s = 0.f;
    for (int d = threadIdx.x; d < D_DIM; d += blockDim.x) {
        float v = C[(size_t)k * D_DIM + d];
        s += v * v;
    }
    __shared__ float red[256];
    red[threadIdx.x] = s;
    __syncthreads();
    for (int st = blockDim.x >> 1; st > 0; st >>= 1) {
        if (threadIdx.x < st) red[threadIdx.x] += red[threadIdx.x + st];
        __syncthreads();
    }
    if (threadIdx.x == 0) mSm[k] = red[0];
}

// ---------------- binary entropy sum over mask (non-temporal stream) ----------------
__device__ __forceinline__ float bent(float p) {
    p = fminf(fmaxf(p, EPSF), 1.0f - EPSF);
    return -(p * logf(p) + (1.0f - p) * logf(1.0f - p));
}
__global__ void k_entropy(const float* __restrict__ p, float* __restrict__ out) {
    const size_t n4 = (size_t)B_ROWS * D_DIM / 4;
    const f4* p4 = (const f4*)p;
    float s = 0.f;
    for (size_t i = blockIdx.x * (size_t)blockDim.x + threadIdx.x; i < n4;
         i += (size_t)gridDim.x * blockDim.x) {
        f4 v = __builtin_nontemporal_load(&p4[i]);   // keep X resident in L2
        s += bent(v.x) + bent(v.y) + bent(v.z) + bent(v.w);
    }
    __shared__ float red[256];
    red[threadIdx.x] = s;
    __syncthreads();
    for (int st = blockDim.x >> 1; st > 0; st >>= 1) {
        if (threadIdx.x < st) red[threadIdx.x] += red[threadIdx.x + st];
        __syncthreads();
    }
    if (threadIdx.x == 0) atomicAdd(out, red[0]);
}

// ---------------- final scalars ----------------
__global__ void k_finalize(const float* __restrict__ counts,
                           const float* __restrict__ sumdiag0, const float* __restrict__ mSm0,
                           const float* __restrict__ ent0,
                           const float* __restrict__ sumdiag1, const float* __restrict__ mSm1,
                           const float* __restrict__ ent1,
                           float* __restrict__ out) {
    if (threadIdx.x != 0 || blockIdx.x != 0) return;
    const float* sd[2] = {sumdiag0, sumdiag1};
    const float* ms[2] = {mSm0, mSm1};
    float sims[2];
    for (int t = 0; t < 2; ++t) {
        float sum = 0.f, nvalid = 0.f;
        for (int k = 0; k < K_CLS; ++k) {
            float n = counts[k];
            float pair_sum = 0.5f * (ms[t][k] - sd[t][k]);
            float n_pairs  = 0.5f * n * (n - 1.0f);
            if (n >= 2.0f) { sum += pair_sum / fmaxf(n_pairs, 1.0f); nvalid += 1.0f; }
        }
        float cs = (nvalid > 0.f) ? (sum / fmaxf(nvalid, 1.0f)) : 0.f;
        sims[t] = -cs;   // sim_loss = -cossim
    }
    float inv_n = 1.0f / ((float)B_ROWS * (float)D_DIM);
    float sp0 = ent0[0] * inv_n;
    float sp1 = ent1[0] * inv_n;
    out[0] = sims[0] + 0.001f * sp0 + sims[1] + 0.001f * sp1;
    out[1] = sims[0];
    out[2] = sims[1];
    out[3] = sp0;
    out[4] = sp1;
}

// ---------------- workspace layout (bytes) ----------------
// [0,32768)          labels (int[8192])
// [32768,33024)      counts (float[64])
// [33024,33536)      sumdiag[2] (float[64] each)
// [33536,34048)      mSm[2]     (float[64] each)
// [34048,34064)      ent[2]     (float each, padded)
// [34304, +64KB)     w[2]       (float[8192] each)
// [99840, +2MB)      C[2]       (float[64*4096] each)

extern "C" void kernel_launch(void* const* d_in, const int* in_sizes, int n_in,
                              void* d_out, int out_size, void* d_ws, size_t ws_size,
                              hipStream_t stream) {
    const float* probs = (const float*)d_in[0];
    const float* acts[2] = { (const float*)d_in[1], (const float*)d_in[2] };
    const float* masks[2] = { (const float*)d_in[3], (const float*)d_in[4] };
    float* out = (float*)d_out;

    char* ws = (char*)d_ws;
    int*   labels  = (int*)(ws + 0);
    float* counts  = (float*)(ws + 32768);
    float* sumdiag[2] = { (float*)(ws + 33024), (float*)(ws + 33024 + 256) };
    float* mSm[2]     = { (float*)(ws + 33536), (float*)(ws + 33536 + 256) };
    float* ent[2]     = { (float*)(ws + 34048), (float*)(ws + 34048 + 8) };
    float* w[2]       = { (float*)(ws + 34304), (float*)(ws + 34304 + 32768) };
    float* C[2]       = { (float*)(ws + 99840), (float*)(ws + 99840 + (size_t)K_CLS * D_DIM * 4) };

    // zero the atomic-accumulated regions (counts..ent, and both centroid buffers)
    hipMemsetAsync(ws + 32768, 0, 34064 - 32768, stream);
    hipMemsetAsync(ws + 99840, 0, 2 * (size_t)K_CLS * D_DIM * 4, stream);

    k_labels<<<(B_ROWS + 255) / 256, 256, 0, stream>>>(probs, labels, counts);

    for (int t = 0; t < 2; ++t) {
        // pass 1: norms (streams X into L2)
        k_row_norms<<<B_ROWS, 256, 0, stream>>>(acts[t], labels, w[t], sumdiag[t]);
        // pass 2: centroid GEMM via WMMA (re-reads X from L2)
        k_centroid_wmma<<<(256 * NSLICE * 32) / 128, 128, 0, stream>>>(acts[t], w[t], labels, C[t]);
        // ||C_k||^2
        k_class_sqnorm<<<K_CLS, 256, 0, stream>>>(C[t], mSm[t]);
        // mask entropy (non-temporal so it doesn't evict the other X tensor)
        k_entropy<<<4096, 256, 0, stream>>>(masks[t], ent[t]);
    }

    k_finalize<<<1, 32, 0, stream>>>(counts, sumdiag[0], mSm[0], ent[0],
                                     sumdiag[1], mSm[1], ent[1], out);
}